// GhostModule_42932493091479
// MI455X (gfx1250) — compile-verified
//
#include <hip/hip_runtime.h>

typedef __attribute__((ext_vector_type(2))) float v2f;
typedef __attribute__((ext_vector_type(8))) float v8f;

#define BN_EPS 1e-5f

// Problem dims (fixed by the reference's setup_inputs)
constexpr int N_  = 4;
constexpr int CIN = 64;
constexpr int H_  = 56;
constexpr int W_  = 56;
constexpr int C1  = 32;          // primary-branch output channels
constexpr int C2  = 32;          // adder-branch output channels
constexpr int K1  = CIN * 9;     // 576  (im2col K for conv)
constexpr int K2  = C1 * 9;      // 288  (patch length for adder)
constexpr int HW  = H_ * W_;     // 3136
constexpr int PIX = N_ * HW;     // 12544 output pixels
constexpr int OUPC = 64;         // output channels in d_out

// ---------------------------------------------------------------------------
// Kernel 0: one-shot weight transpose into workspace.
// Bt[tap][c][o] = wp[o][c*9 + tap]   (9 x 64 x 32 floats = 73728 B)
// Makes every WMMA B-fragment load lane-consecutive (coalesced) and lets the
// conv inner loop walk B linearly in c.
// ---------------------------------------------------------------------------
__global__ __launch_bounds__(256) void ghost_wt_transpose_kernel(
    const float* __restrict__ wp,    // [32][576]
    float* __restrict__ Bt)          // [9][64][32]
{
    const int i = blockIdx.x * blockDim.x + threadIdx.x;  // 18432 total
    const int o   = i & 31;
    const int t   = i >> 5;
    const int c   = t & 63;
    const int tap = t >> 6;
    Bt[i] = wp[o * K1 + c * 9 + tap];
}

// ---------------------------------------------------------------------------
// Kernel 1: 3x3 conv (64->32) as 9 tap-decomposed rank-64 GEMMs on
// V_WMMA_F32_16X16X4_F32, fused BN + ReLU. One wave computes a
// 16-pixel x 32-channel tile. Writes x1 into d_out channels [0,32).
// ---------------------------------------------------------------------------
__global__ __launch_bounds__(256) void ghost_conv_wmma_kernel(
    const float* __restrict__ x,      // [4,64,56,56]
    const float* __restrict__ Bt,     // [9][64][32] transposed weights
    const float* __restrict__ gamma,
    const float* __restrict__ beta,
    const float* __restrict__ mean,
    const float* __restrict__ var,
    float* __restrict__ out)          // [4,64,56,56]
{
    const int lane = threadIdx.x & 31;
    const int wave = threadIdx.x >> 5;
    const int tile = blockIdx.x * (blockDim.x >> 5) + wave;  // 784 tiles of 16 px
    const int m    = lane & 15;   // pixel-in-tile (A rows) / channel col (B cols)
    const int khi  = lane >> 4;   // selects K pair {k0,k0+1} vs {k0+2,k0+3}

    // Pixel coordinates for this lane's A row (HW is a multiple of 16, so a
    // tile never crosses an image boundary -> n,h base uniform per tile).
    const int p  = tile * 16 + m;
    const int n  = p / HW;
    const int hw = p % HW;
    const int h  = hw / W_;
    const int w  = hw % W_;
    const float* xn = x + n * CIN * HW;

    v8f acc0 = {0.f, 0.f, 0.f, 0.f, 0.f, 0.f, 0.f, 0.f};  // channels [0,16)
    v8f acc1 = {0.f, 0.f, 0.f, 0.f, 0.f, 0.f, 0.f, 0.f};  // channels [16,32)

    for (int tap = 0; tap < 9; ++tap) {
        const int kh = tap / 3;
        const int kw = tap - kh * 3;
        const int hh = h + kh - 1;
        const int ww = w + kw - 1;
        const bool ib = ((unsigned)hh < (unsigned)H_) &&
                        ((unsigned)ww < (unsigned)W_);
        const float* xp  = xn + (hh * W_ + ww);           // deref only when ib
        const float* bt0 = Bt + (tap * CIN) * C1 + m;     // cols [0,16)
        const float* bt1 = bt0 + 16;                      // cols [16,32)

#pragma unroll 4
        for (int c0 = 0; c0 < CIN; c0 += 4) {
            const int ca = c0 + 2 * khi;
            v2f a;
            a[0] = ib ? xp[ ca      * HW] : 0.f;
            a[1] = ib ? xp[(ca + 1) * HW] : 0.f;
            v2f b0, b1;
            b0[0] = bt0[ ca      * C1];
            b0[1] = bt0[(ca + 1) * C1];
            b1[0] = bt1[ ca      * C1];
            b1[1] = bt1[(ca + 1) * C1];
            acc0 = __builtin_amdgcn_wmma_f32_16x16x4_f32(
                false, a, false, b0, (short)0, acc0, false, false);
            acc1 = __builtin_amdgcn_wmma_f32_16x16x4_f32(
                false, a, false, b1, (short)0, acc1, false, false);
        }
    }

    // Epilogue: BN + ReLU + scatter D. D layout: VGPR r holds row (r + 8*khi),
    // col = lane&15. Row pixel = tile*16 + r + 8*khi, same image as p.
    const int ch0 = m;
    const int ch1 = m + 16;
    const float inv0 = gamma[ch0] / sqrtf(var[ch0] + BN_EPS);
    const float inv1 = gamma[ch1] / sqrtf(var[ch1] + BN_EPS);
    const float mu0 = mean[ch0], mu1 = mean[ch1];
    const float be0 = beta[ch0], be1 = beta[ch1];
    float* out0 = out + (n * OUPC + ch0) * HW;
    float* out1 = out + (n * OUPC + ch1) * HW;
    const int hwbase = hw - m + 8 * khi;  // tile-start pixel within image (+half)

#pragma unroll
    for (int r = 0; r < 8; ++r) {
        const int hw2 = hwbase + r;
        float y0 = (acc0[r] - mu0) * inv0 + be0;
        float y1 = (acc1[r] - mu1) * inv1 + be1;
        out0[hw2] = fmaxf(y0, 0.f);
        out1[hw2] = fmaxf(y1, 0.f);
    }
}

// ---------------------------------------------------------------------------
// Kernel 2: AdderNet branch on x1 (= d_out channels [0,32)).
// out2[n,o,h,w] = -sum_k |patch(x1)[k] - w_adder[o][k]|, then BN + ReLU.
// Weights staged in LDS as [k][o] (lane-consecutive o -> conflict-free).
// One wave = one pixel x 32 output channels; patch loads broadcast.
// ---------------------------------------------------------------------------
__global__ __launch_bounds__(256) void ghost_adder_kernel(
    const float* __restrict__ wa,     // [32,32,3,3] = [o][k]
    const float* __restrict__ gamma,
    const float* __restrict__ beta,
    const float* __restrict__ mean,
    const float* __restrict__ var,
    float* __restrict__ out)          // reads ch [0,32), writes ch [32,64)
{
    __shared__ float Wlds[K2 * C2];   // 288*32 floats = 36 KB
    for (int i = threadIdx.x; i < K2 * C2; i += blockDim.x) {
        const int o = i & 31;
        const int k = i >> 5;
        Wlds[k * C2 + o] = wa[o * K2 + k];
    }
    __syncthreads();

    const int o  = threadIdx.x & 31;          // output channel
    const int pg = threadIdx.x >> 5;          // pixel group within block (0..7)
    const int p  = blockIdx.x * 8 + pg;       // global pixel
    const int n  = p / HW;
    const int hw = p % HW;
    const int h  = hw / W_;
    const int w  = hw % W_;
    const float* x1 = out + n * OUPC * HW;    // channels [0,32) hold x1

    float acc = 0.f;
#pragma unroll
    for (int kh = 0; kh < 3; ++kh) {
        const int hh = h + kh - 1;
#pragma unroll
        for (int kw = 0; kw < 3; ++kw) {
            const int ww  = w + kw - 1;
            const bool ib = ((unsigned)hh < (unsigned)H_) &&
                            ((unsigned)ww < (unsigned)W_);
            const int off = hh * W_ + ww;
            const int kb  = kh * 3 + kw;
            for (int c = 0; c < C1; ++c) {
                const float xv = ib ? x1[c * HW + off] : 0.f;
                const float wv = Wlds[(c * 9 + kb) * C2 + o];
                acc -= fabsf(xv - wv);
            }
        }
    }

    const float inv = gamma[o] / sqrtf(var[o] + BN_EPS);
    const float y   = (acc - mean[o]) * inv + beta[o];
    out[(n * OUPC + C1 + o) * HW + hw] = fmaxf(y, 0.f);
}

// ---------------------------------------------------------------------------
extern "C" void kernel_launch(void* const* d_in, const int* in_sizes, int n_in,
                              void* d_out, int out_size, void* d_ws, size_t ws_size,
                              hipStream_t stream) {
    (void)in_sizes; (void)n_in; (void)out_size; (void)ws_size;

    const float* x   = (const float*)d_in[0];
    const float* wp  = (const float*)d_in[1];
    const float* g1  = (const float*)d_in[2];
    const float* b1  = (const float*)d_in[3];
    const float* m1  = (const float*)d_in[4];
    const float* v1  = (const float*)d_in[5];
    const float* wa  = (const float*)d_in[6];
    const float* g2  = (const float*)d_in[7];
    const float* b2  = (const float*)d_in[8];
    const float* m2  = (const float*)d_in[9];
    const float* v2  = (const float*)d_in[10];
    float* out = (float*)d_out;
    float* Bt  = (float*)d_ws;        // 9*64*32 floats = 73728 B of scratch

    // Transpose conv weights: [32][576] -> [9][64][32]
    ghost_wt_transpose_kernel<<<(9 * CIN * C1) / 256, 256, 0, stream>>>(wp, Bt);

    // 98 blocks * 8 waves * 16 pixels = 12544 = N*H*W (exact, no partial waves
    // -> EXEC all-1s for every WMMA).
    ghost_conv_wmma_kernel<<<PIX / (8 * 16), 256, 0, stream>>>(
        x, Bt, g1, b1, m1, v1, out);

    // 1568 blocks * 8 pixels = 12544; kernel reads x1 from d_out ch[0,32).
    ghost_adder_kernel<<<PIX / 8, 256, 0, stream>>>(
        wa, g2, b2, m2, v2, out);
}